// SupConLoss_19095424598569
// MI455X (gfx1250) — compile-verified
//
#include <hip/hip_runtime.h>

// SupConLoss fused kernel for gfx1250 (MI455X).
// sim = (C C^T)/T via V_WMMA_F32_16X16X4_F32 (exact fp32), fused with online
// (base-2) softmax + positive-mask accumulation.
// Column tiles are staged in LDS once per workgroup with
// GLOBAL_LOAD_ASYNC_TO_LDS_B128 (ASYNCcnt) and double-buffered, so the 8 waves
// of a block share one 8 KB fetch per 16x128 tile instead of fetching it 8x.
//
// contrast row n -> features[b = n%4096, v = n/4096, :], element (b,v,k) at ((b*2)+v)*128 + k.

typedef __attribute__((ext_vector_type(2))) float v2f;
typedef __attribute__((ext_vector_type(8))) float v8f;

#define BSZ       4096
#define NTOT      8192          // N = bsz * n_views
#define DIM       128
#define KCHUNKS   32            // 128 / 4
#define NBLK      512           // NTOT / 16 column (and row) tiles
#define INV_T     (1.0f / 0.07f)
#define LOG2E     1.4426950408889634f
#define LN2       0.6931471805599453f
#define BSTRIDE   132           // LDS row stride in floats: 528 B, 16B-aligned, conflict-free

__device__ __forceinline__ const float* crow_ptr(const float* feat, int n) {
    // contrast_feature row n base pointer
    return feat + (((n & (BSZ - 1)) * 2 + (n >> 12)) << 7);
}

__device__ __forceinline__ void async_b128_pair(const float* gsrc, unsigned ldsOff) {
    // 32 bytes: two async B128 copies; inst offset applies to both LDS and global addr.
    asm volatile("global_load_async_to_lds_b128 %0, %1, off"
                 :: "v"(ldsOff), "v"(gsrc) : "memory");
    asm volatile("global_load_async_to_lds_b128 %0, %1, off offset:16"
                 :: "v"(ldsOff), "v"(gsrc) : "memory");
}

__device__ __forceinline__ void wait_asynccnt0() {
    asm volatile("s_wait_asynccnt 0x0" ::: "memory");
}

__global__ __launch_bounds__(256) void supcon_main(const float* __restrict__ feat,
                                                   const int*  __restrict__ labels,
                                                   float*      __restrict__ partial) {
    __shared__ int   slab[BSZ];                  // 16 KB labels
    __shared__ float btile[2][16 * BSTRIDE];     // 2 x 8.25 KB double-buffered column tile

    const int tid = threadIdx.x;
    #pragma unroll
    for (int i = 0; i < BSZ / 256; ++i)
        slab[tid + i * 256] = labels[tid + i * 256];

    const int wave = tid >> 5;        // 0..7
    const int lane = tid & 31;
    const int h    = lane >> 4;       // half of wave: 0 or 1 (K-pair / row+8 select)
    const int l    = lane & 15;

    const int I = blockIdx.x * 8 + wave;          // row block 0..511

    // producer mapping: each thread copies 32 B of the 16x128 tile
    const int prow = tid >> 4;        // 0..15: row within tile
    const int pseg = tid & 15;        // 0..15: 8-float segment within row
    const unsigned lds0 = (unsigned)(uintptr_t)(&btile[0][prow * BSTRIDE + pseg * 8]);
    const unsigned lds1 = (unsigned)(uintptr_t)(&btile[1][prow * BSTRIDE + pseg * 8]);

    // ---- preload A fragments for this wave's 16 rows (all K) ----
    // A layout: lane l (each half) holds M=l; half h supplies K = kc*4 + 2h + {0,1}
    const float* aptr = crow_ptr(feat, I * 16 + l) + h * 2;
    v2f afrag[KCHUNKS];
    #pragma unroll
    for (int kc = 0; kc < KCHUNKS; ++kc)
        afrag[kc] = *(const v2f*)(aptr + kc * 4);

    // the 8 output rows this lane sees: row = 16I + r + 8h  (D-matrix layout)
    int rlab[8], rglob[8];

    // prefetch tile 0 into buffer 0 (async), overlapping with A preload above
    async_b128_pair(crow_ptr(feat, 0 * 16 + prow) + pseg * 8, lds0);

    #pragma unroll
    for (int r = 0; r < 8; ++r) {
        int row = I * 16 + r + 8 * h;
        rglob[r] = row;
    }

    // online-softmax state per lane (base-2 domain) + masked sums
    float m[8], z[8], S[8], cnt[8];
    #pragma unroll
    for (int r = 0; r < 8; ++r) { m[r] = -3.0e38f; z[r] = 0.f; S[r] = 0.f; cnt[r] = 0.f; }

    const float scale = INV_T * LOG2E;   // sim -> log2 domain

    wait_asynccnt0();
    __syncthreads();                     // slab + tile 0 visible to whole block

    #pragma unroll
    for (int r = 0; r < 8; ++r)
        rlab[r] = slab[rglob[r] & (BSZ - 1)];

    for (int J = 0; J < NBLK; ++J) {
        const int cur = J & 1;
        // prefetch next column tile into the other buffer while we compute
        if (J + 1 < NBLK) {
            async_b128_pair(crow_ptr(feat, (J + 1) * 16 + prow) + pseg * 8,
                            cur ? lds0 : lds1);
        }

        const int col  = J * 16 + l;     // this lane's column in the tile (D layout)
        const int clab = slab[col & (BSZ - 1)];
        // B layout mirrors A: lane l holds N=col, half h supplies K = kc*4 + 2h + {0,1}
        const float* bp = &btile[cur][l * BSTRIDE + h * 2];

        v8f c0 = {0.f, 0.f, 0.f, 0.f, 0.f, 0.f, 0.f, 0.f};
        v8f c1 = {0.f, 0.f, 0.f, 0.f, 0.f, 0.f, 0.f, 0.f};
        #pragma unroll
        for (int kc = 0; kc < KCHUNKS; kc += 2) {
            v2f b0 = *(const v2f*)(bp + kc * 4);
            v2f b1 = *(const v2f*)(bp + kc * 4 + 4);
            c0 = __builtin_amdgcn_wmma_f32_16x16x4_f32(false, afrag[kc],     false, b0,
                                                       (short)0, c0, false, false);
            c1 = __builtin_amdgcn_wmma_f32_16x16x4_f32(false, afrag[kc + 1], false, b1,
                                                       (short)0, c1, false, false);
        }

        #pragma unroll
        for (int r = 0; r < 8; ++r) {
            float t = (c0[r] + c1[r]) * scale;            // sim * log2(e)
            bool pos = (clab == rlab[r]) & (col != rglob[r]);
            if (pos) { S[r] += t; cnt[r] += 1.0f; }
            float mo = m[r];
            float mn = fmaxf(mo, t);
            z[r] = z[r] * __builtin_amdgcn_exp2f(mo - mn)
                 + __builtin_amdgcn_exp2f(t - mn);
            m[r] = mn;
        }

        wait_asynccnt0();    // my prefetch for J+1 has landed
        __syncthreads();     // whole block done reading buffer `cur`
    }

    // ---- reduce the 16 lanes of each half (columns of each row) ----
    float wavesum = 0.f;
    #pragma unroll
    for (int r = 0; r < 8; ++r) {
        float mr = m[r], zr = z[r], Sr = S[r], cr = cnt[r];
        #pragma unroll
        for (int off = 8; off > 0; off >>= 1) {
            float mo = __shfl_xor(mr, off, 16);
            float zo = __shfl_xor(zr, off, 16);
            float So = __shfl_xor(Sr, off, 16);
            float co = __shfl_xor(cr, off, 16);
            float mn = fmaxf(mr, mo);
            zr = zr * __builtin_amdgcn_exp2f(mr - mn)
               + zo * __builtin_amdgcn_exp2f(mo - mn);
            mr = mn;
            Sr += So;
            cr += co;
        }
        // mean_log_prob_pos = ln2 * (S/cnt - m - log2(z));  loss_i = -that
        float lp = LN2 * (Sr / cr - mr - __builtin_amdgcn_logf(zr));
        wavesum += -lp;
    }
    // combine the two halves (rows 0-7 and 8-15 of the block)
    wavesum += __shfl_xor(wavesum, 16, 32);
    if (lane == 0) partial[I] = wavesum;
}

__global__ __launch_bounds__(256) void supcon_finish(const float* __restrict__ partial,
                                                     float* __restrict__ out) {
    __shared__ float red[256];
    int tid = threadIdx.x;
    red[tid] = partial[tid] + partial[tid + 256];
    __syncthreads();
    #pragma unroll
    for (int s = 128; s > 0; s >>= 1) {
        if (tid < s) red[tid] += red[tid + s];
        __syncthreads();
    }
    if (tid == 0) out[0] = red[0] * (1.0f / (float)NTOT);
}

extern "C" void kernel_launch(void* const* d_in, const int* in_sizes, int n_in,
                              void* d_out, int out_size, void* d_ws, size_t ws_size,
                              hipStream_t stream) {
    (void)in_sizes; (void)n_in; (void)out_size; (void)ws_size;
    const float* feat   = (const float*)d_in[0];   // [4096, 2, 128] f32
    const int*   labels = (const int*)d_in[1];     // [4096] i32
    float* out     = (float*)d_out;                // scalar f32
    float* partial = (float*)d_ws;                 // 512 floats of scratch

    supcon_main<<<dim3(64), dim3(256), 0, stream>>>(feat, labels, partial);
    supcon_finish<<<dim3(1), dim3(256), 0, stream>>>(partial, out);
}